// LGS_19911468384777
// MI455X (gfx1250) — compile-verified
//
#include <hip/hip_runtime.h>

// ---------------- problem constants ----------------
#define HH   384
#define WW   1280
#define CC   3
#define NIMG 16            // 8 per input batch
#define HWSZ (HH*WW)       // 491520
#define CHW  (CC*HWSZ)
#define NH   48            // patch rows
#define NW   160           // patch cols
#define PAD  4
#define THRT 0.15f
#define SSC  2.3f
#define EPSG 1e-6f

typedef float v2f __attribute__((ext_vector_type(2)));
typedef float v8f __attribute__((ext_vector_type(8)));

__device__ __forceinline__ int refl(int u, int n) {
    u = (u < 0) ? -u : u;
    return (u >= n) ? (2 * n - 2 - u) : u;
}

// ---------------- init: per-image min/max seeds ----------------
__global__ void lgs_init(unsigned* __restrict__ minb, unsigned* __restrict__ maxb) {
    int t = threadIdx.x;
    if (t < NIMG) {
        minb[t] = 0x7F800000u; // +inf
        maxb[t] = 0u;          // 0.0f (G > 0 always)
    }
}

// ---------------- pass 1: Sobel magnitude + per-image min/max ----------------
__global__ void lgs_pass1(const float* __restrict__ I1, const float* __restrict__ I2,
                          float* __restrict__ G,
                          unsigned* __restrict__ minb, unsigned* __restrict__ maxb) {
    __shared__ float smin[256];
    __shared__ float smax[256];

    int gid = blockIdx.x * 256 + threadIdx.x;   // exact: NIMG*HWSZ threads
    int n = gid / HWSZ;
    int p = gid - n * HWSZ;
    int y = p / WW;
    int x = p - y * WW;

    const float* I = (n < 8) ? (I1 + (size_t)n * CHW) : (I2 + (size_t)(n - 8) * CHW);

    int ym = (y == 0)      ? 1      : y - 1;
    int yp = (y == HH - 1) ? HH - 2 : y + 1;
    int xm = (x == 0)      ? 1      : x - 1;
    int xp = (x == WW - 1) ? WW - 2 : x + 1;

    float acc = EPSG;
#pragma unroll
    for (int c = 0; c < CC; ++c) {
        const float* B = I + (size_t)c * HWSZ;
        float a00 = B[ym * WW + xm], a01 = B[ym * WW + x], a02 = B[ym * WW + xp];
        float a10 = B[y  * WW + xm],                       a12 = B[y  * WW + xp];
        float a20 = B[yp * WW + xm], a21 = B[yp * WW + x], a22 = B[yp * WW + xp];
        float dx = 0.125f * (a00 - a02) + 0.25f * (a10 - a12) + 0.125f * (a20 - a22);
        float dy = 0.125f * (a00 - a20) + 0.25f * (a01 - a21) + 0.125f * (a02 - a22);
        acc += dx * dx + dy * dy;
    }
    float g = sqrtf(acc);
    G[gid] = g;

    // block-level min/max tree, then one atomic per block (block lies in one image)
    int t = threadIdx.x;
    smin[t] = g;
    smax[t] = g;
    __syncthreads();
    for (int off = 128; off > 0; off >>= 1) {
        if (t < off) {
            smin[t] = fminf(smin[t], smin[t + off]);
            smax[t] = fmaxf(smax[t], smax[t + off]);
        }
        __syncthreads();
    }
    if (t == 0) {
        atomicMin(&minb[n], __float_as_uint(smin[0]));  // valid: G > 0
        atomicMax(&maxb[n], __float_as_uint(smax[0]));
    }
}

// ---------------- pass 2: 16x16 patch means of normalized G (one wave / patch) ----
__global__ void lgs_pass2(const float* __restrict__ G,
                          const unsigned* __restrict__ minb, const unsigned* __restrict__ maxb,
                          float* __restrict__ Bmean) {
    int wave = (blockIdx.x * blockDim.x + threadIdx.x) >> 5;  // exact: NIMG*NH*NW waves
    int lane = threadIdx.x & 31;

    int n  = wave / (NH * NW);
    int r  = wave - n * (NH * NW);
    int bi = r / NW;
    int bj = r - bi * NW;

    float gmin = __uint_as_float(minb[n]);
    float gmax = __uint_as_float(maxb[n]);
    float inv  = 1.0f / (gmax - gmin);

    const float* Gi = G + (size_t)n * HWSZ;

    int m    = lane & 15;      // A-matrix row owned by this lane
    int half = lane >> 4;      // lanes 16-31 supply K=+2,+3 columns
    int gy   = refl(bi * 8 + m - PAD, HH);
    const float* Grow = Gi + (size_t)gy * WW;

#if __has_builtin(__builtin_amdgcn_wmma_f32_16x16x4_f32)
    // D = A(16x4,f32) x ones(4x16) + C ; 4 steps cover K=0..15.
    // D[m,*] = row-sum of patch row m (f32 accumulate, no precision loss).
    v2f bones; bones.x = 1.0f; bones.y = 1.0f;
    v8f cacc = {};
#pragma unroll
    for (int k = 0; k < 16; k += 4) {
        int c0  = bj * 8 + k + 2 * half - PAD;
        int gx0 = refl(c0,     WW);
        int gx1 = refl(c0 + 1, WW);
        v2f a;
        a.x = (Grow[gx0] - gmin) * inv;   // A[m, k + 2*half]
        a.y = (Grow[gx1] - gmin) * inv;   // A[m, k + 2*half + 1]
        cacc = __builtin_amdgcn_wmma_f32_16x16x4_f32(
            /*neg_a=*/false, a, /*neg_b=*/false, bones,
            /*c_mod=*/(short)0, cacc, /*reuse_a=*/false, /*reuse_b=*/false);
    }
    // lane<16 holds D rows 0..7, lane>=16 holds rows 8..15 (same value along N)
    float part = cacc[0] + cacc[1] + cacc[2] + cacc[3] +
                 cacc[4] + cacc[5] + cacc[6] + cacc[7];
    part += __shfl_xor(part, 16, 32);
    if (lane == 0) Bmean[wave] = part * (1.0f / 256.0f);
#else
    // fallback: 8 elems per lane + full wave32 shuffle reduction
    float s = 0.0f;
#pragma unroll
    for (int t = 0; t < 8; ++t) {
        int e   = lane + 32 * t;
        int row = e >> 4, col = e & 15;
        int gy2 = refl(bi * 8 + row - PAD, HH);
        int gx  = refl(bj * 8 + col - PAD, WW);
        s += (Gi[(size_t)gy2 * WW + gx] - gmin) * inv;
    }
    for (int off = 16; off > 0; off >>= 1) s += __shfl_xor(s, off, 32);
    if (lane == 0) Bmean[wave] = s * (1.0f / 256.0f);
#endif
}

// ---------------- pass 3: gate + modulate, streamed output ----------------
__global__ void lgs_pass3(const float* __restrict__ I1, const float* __restrict__ I2,
                          const float* __restrict__ G,
                          const unsigned* __restrict__ minb, const unsigned* __restrict__ maxb,
                          const float* __restrict__ Bmean,
                          float* __restrict__ Out) {
    int gid = blockIdx.x * 256 + threadIdx.x;   // exact: NIMG*HWSZ threads
    int n = gid / HWSZ;
    int p = gid - n * HWSZ;
    int y = p / WW;
    int x = p - y * WW;

    float gmin = __uint_as_float(minb[n]);
    float gmax = __uint_as_float(maxb[n]);
    float gn   = (G[gid] - gmin) / (gmax - gmin);

    // covering patches of padded coordinate (y+4, x+4); interior pixels have 2x2
    int py = y + PAD, px = x + PAD;
    int bi0 = max(0, (py - 8) >> 3), bi1 = min(NH - 1, py >> 3);
    int bj0 = max(0, (px - 8) >> 3), bj1 = min(NW - 1, px >> 3);

    const float* Bm = Bmean + (size_t)n * (NH * NW);
    bool act = false;
    for (int bi = bi0; bi <= bi1; ++bi)
        for (int bj = bj0; bj <= bj1; ++bj)
            act = act || (Bm[bi * NW + bj] > THRT);

    float fm  = act ? gn : 0.0f;
    float fac = 1.0f - fminf(fmaxf(SSC * fm, 0.0f), 1.0f);

    const float* I = (n < 8) ? (I1 + (size_t)n * CHW) : (I2 + (size_t)(n - 8) * CHW);
    float* O = Out + (size_t)n * CHW;   // d_out is [lgs(I1) flat ; lgs(I2) flat]
#pragma unroll
    for (int c = 0; c < CC; ++c) {
        size_t off = (size_t)c * HWSZ + p;
        __builtin_nontemporal_store(I[off] * fac, &O[off]);
    }
}

extern "C" void kernel_launch(void* const* d_in, const int* in_sizes, int n_in,
                              void* d_out, int out_size, void* d_ws, size_t ws_size,
                              hipStream_t stream) {
    const float* I1 = (const float*)d_in[0];
    const float* I2 = (const float*)d_in[1];
    float* Out = (float*)d_out;

    // workspace layout
    float*    G     = (float*)d_ws;                         // NIMG*HWSZ f32 (31.5 MB)
    unsigned* minb  = (unsigned*)(G + (size_t)NIMG * HWSZ); // NIMG
    unsigned* maxb  = minb + NIMG;                          // NIMG
    float*    Bmean = (float*)(maxb + NIMG);                // NIMG*NH*NW f32

    lgs_init<<<1, 32, 0, stream>>>(minb, maxb);

    int pixBlocks = (NIMG * HWSZ) / 256;                    // 30720, exact
    lgs_pass1<<<pixBlocks, 256, 0, stream>>>(I1, I2, G, minb, maxb);

    int waveBlocks = (NIMG * NH * NW * 32) / 256;           // 15360, exact
    lgs_pass2<<<waveBlocks, 256, 0, stream>>>(G, minb, maxb, Bmean);

    lgs_pass3<<<pixBlocks, 256, 0, stream>>>(I1, I2, G, minb, maxb, Bmean, Out);
}